// WSCL_SDA_Loss_34205119545437
// MI455X (gfx1250) — compile-verified
//
#include <hip/hip_runtime.h>
#include <hip/hip_bf16.h>
#include <stdint.h>
#include <float.h>

// ---------------------------------------------------------------------------
// WSCL supervised-contrastive loss + hard-positive mining, MI455X (gfx1250).
// Memory-bound masked reductions: stream logits/CID/TID tiles through LDS with
// CDNA5 async global->LDS copies (ASYNCcnt / s_wait_asynccnt), double-buffered,
// single-exp online softmax, wave32 shuffle reductions. No matmul -> no WMMA.
// ---------------------------------------------------------------------------

#define TEMPERATURE      0.07f
#define BASE_TEMPERATURE 0.07f
#define INV_TEMP         (1.0f / TEMPERATURE)

#define TEAM     8      // blocks per anchor (split of N)
#define NTHREADS 256    // 8 waves of 32
#define TILE     1024   // floats per staged tile; TILE == 4*NTHREADS

#define NEG_SENTINEL (-FLT_MAX)   // "empty" softmax max (finite: no NaN paths)

// ---- CDNA5 async global->LDS path (guarded; falls back to plain loads) ----
#ifndef USE_ASYNC
#if defined(__HIP_DEVICE_COMPILE__) && \
    __has_builtin(__builtin_amdgcn_global_load_async_to_lds_b128) && \
    __has_builtin(__builtin_amdgcn_s_wait_asynccnt)
#define USE_ASYNC 1
#else
#define USE_ASYNC 0
#endif
#endif

#if USE_ASYNC
typedef int v4i_async __attribute__((vector_size(16)));
__device__ __forceinline__ void async_copy16(const void* gsrc, void* ldst) {
  // Builtin signature: (v4i AS(1)*, v4i AS(3)*, imm offset, imm cpol).
  // generic->AS1 is numeric identity; generic LDS addr low 32 bits == LDS
  // offset per the flat-aperture mapping.
  __builtin_amdgcn_global_load_async_to_lds_b128(
      (__attribute__((address_space(1))) v4i_async*)(uintptr_t)gsrc,
      (__attribute__((address_space(3))) v4i_async*)(uint32_t)(uintptr_t)ldst,
      /*offset=*/0, /*cpol=*/0);
}
#define ASYNC_WAIT(n) __builtin_amdgcn_s_wait_asynccnt(n)
#else
#define ASYNC_WAIT(n) do { } while (0)
#endif

// merge two online-softmax states (m,s) <- (m,s) (+) (om,os).
// Sentinel-empty states use m == -FLT_MAX, s == 0. Single exp, no NaN:
// d may overflow to +/-inf -> exp(-|d|) == 0 picks the surviving side.
__device__ __forceinline__ void lse_merge(float& m, float& s, float om, float os) {
  float d = om - m;
  float e = __expf(-fabsf(d));
  if (d > 0.f) { s = fmaf(s, e, os); m = om; }
  else         { s = fmaf(os, e, s); }
}

// ---------------------------------------------------------------------------
// Kernel 1: per-(anchor, split) partial reductions over j in [jbeg, jend)
// ---------------------------------------------------------------------------
__global__ __launch_bounds__(NTHREADS) void wscl_partial(
    const float* __restrict__ logits,    // [B, N]
    const int*   __restrict__ memCID,    // [N]
    const int*   __restrict__ memTID,    // [N]
    const int*   __restrict__ camids,    // [B]
    const int*   __restrict__ trackids,  // [B]
    int N,
    float* __restrict__ ws_m,  float* __restrict__ ws_s,
    float* __restrict__ ws_ps, float* __restrict__ ws_pc,
    float* __restrict__ ws_mv, int*   __restrict__ ws_mi) {
  const int b   = blockIdx.x / TEAM;
  const int sp  = blockIdx.x % TEAM;
  const int tid = threadIdx.x;
  const int cam = camids[b];
  const int trk = trackids[b];
  const float* lrow = logits + (size_t)b * (size_t)N;

  const int chunk = (N + TEAM - 1) / TEAM;
  const int jbeg  = sp * chunk;
  const int jend  = (jbeg + chunk < N) ? (jbeg + chunk) : N;
  const int span  = (jend > jbeg) ? (jend - jbeg) : 0;
  const int nFull = span / TILE;

  __shared__ __align__(16) float sh_log[2][TILE];
  __shared__ __align__(16) int   sh_cid[2][TILE];
  __shared__ __align__(16) int   sh_tid[2][TILE];

  float m = NEG_SENTINEL, s = 0.f; // online softmax over camera slice of adc
  float psum = 0.f, pcnt = 0.f;    // sum(adc) and count over positives
  float mv = FLT_MAX;              // min raw logit over positives
  int   mi = 0x7fffffff;           // its (first) index

  const int k4 = tid * 4;          // TILE == 4*NTHREADS: one b128/array/thread

  auto issue = [&](int t, int bufi) {
    const int j0 = jbeg + t * TILE;
#if USE_ASYNC
    async_copy16(&lrow[j0 + k4],   &sh_log[bufi][k4]);
    async_copy16(&memCID[j0 + k4], &sh_cid[bufi][k4]);
    async_copy16(&memTID[j0 + k4], &sh_tid[bufi][k4]);
#else
    __builtin_prefetch(&lrow[j0 + k4], 0, 0);
    *(float4*)(&sh_log[bufi][k4]) = *(const float4*)(&lrow[j0 + k4]);
    *(int4*)(&sh_cid[bufi][k4])   = *(const int4*)(&memCID[j0 + k4]);
    *(int4*)(&sh_tid[bufi][k4])   = *(const int4*)(&memTID[j0 + k4]);
#endif
  };

  if (nFull > 0) issue(0, 0);
  for (int t = 0; t < nFull; ++t) {
    const int  bufi    = t & 1;
    const bool hasNext = (t + 1) < nFull;
    if (hasNext) {                 // prefetch next tile, then drain current
      issue(t + 1, bufi ^ 1);      // 3 async ops/wave now outstanding for t+1
      ASYNC_WAIT(3);               // tile t's 3 ops (older, in-order) are done
    } else {
      ASYNC_WAIT(0);
    }
    __syncthreads();               // tile t visible to all 8 waves

    // consume this thread's contiguous 4 elements via 3x ds_load_b128
    const float4 lv4 = *(const float4*)&sh_log[bufi][k4];
    const int4   c4  = *(const int4*)&sh_cid[bufi][k4];
    const int4   t4  = *(const int4*)&sh_tid[bufi][k4];
    const float lvs[4] = {lv4.x, lv4.y, lv4.z, lv4.w};
    const int   cs[4]  = {c4.x,  c4.y,  c4.z,  c4.w};
    const int   ts[4]  = {t4.x,  t4.y,  t4.z,  t4.w};
    const int j0 = jbeg + t * TILE;
#pragma unroll
    for (int e = 0; e < 4; ++e) {
      if (cs[e] == cam) {
        const float lv = lvs[e];
        const float v  = lv * INV_TEMP;
        const float d  = v - m;
        const float ex = __expf(-fabsf(d));
        if (d > 0.f) { s = fmaf(s, ex, 1.f); m = v; }
        else         { s += ex; }
        if (ts[e] == trk) {
          psum += v;
          pcnt += 1.f;
          const int j = j0 + k4 + e;
          if (lv < mv || (lv == mv && j < mi)) { mv = lv; mi = j; }
        }
      }
    }
    __syncthreads();               // buffer safe to overwrite at t+2
  }

  // tail (span % TILE) straight from global (L2-resident)
  for (int j = jbeg + nFull * TILE + tid; j < jend; j += NTHREADS) {
    const float lv = lrow[j];
    if (memCID[j] == cam) {
      const float v  = lv * INV_TEMP;
      const float d  = v - m;
      const float ex = __expf(-fabsf(d));
      if (d > 0.f) { s = fmaf(s, ex, 1.f); m = v; }
      else         { s += ex; }
      if (memTID[j] == trk) {
        psum += v;
        pcnt += 1.f;
        if (lv < mv || (lv == mv && j < mi)) { mv = lv; mi = j; }
      }
    }
  }

  // -------- wave32 reduction (5 xor steps) --------
#pragma unroll
  for (int off = 16; off > 0; off >>= 1) {
    float om  = __shfl_xor(m,    off, 32);
    float os  = __shfl_xor(s,    off, 32);
    float ops = __shfl_xor(psum, off, 32);
    float opc = __shfl_xor(pcnt, off, 32);
    float omv = __shfl_xor(mv,   off, 32);
    int   omi = __shfl_xor(mi,   off, 32);
    lse_merge(m, s, om, os);
    psum += ops;
    pcnt += opc;
    if (omv < mv || (omv == mv && omi < mi)) { mv = omv; mi = omi; }
  }

  // -------- cross-wave (8 waves) via LDS --------
  __shared__ float r_m[8], r_s[8], r_ps[8], r_pc[8], r_mv[8];
  __shared__ int   r_mi[8];
  const int wave = tid >> 5, lane = tid & 31;
  if (lane == 0) {
    r_m[wave] = m; r_s[wave] = s; r_ps[wave] = psum;
    r_pc[wave] = pcnt; r_mv[wave] = mv; r_mi[wave] = mi;
  }
  __syncthreads();
  if (wave == 0) {
    if (lane < 8) {
      m = r_m[lane]; s = r_s[lane]; psum = r_ps[lane];
      pcnt = r_pc[lane]; mv = r_mv[lane]; mi = r_mi[lane];
    } else {
      m = NEG_SENTINEL; s = 0.f; psum = 0.f; pcnt = 0.f;
      mv = FLT_MAX; mi = 0x7fffffff;
    }
#pragma unroll
    for (int off = 16; off > 0; off >>= 1) {
      float om  = __shfl_xor(m,    off, 32);
      float os  = __shfl_xor(s,    off, 32);
      float ops = __shfl_xor(psum, off, 32);
      float opc = __shfl_xor(pcnt, off, 32);
      float omv = __shfl_xor(mv,   off, 32);
      int   omi = __shfl_xor(mi,   off, 32);
      lse_merge(m, s, om, os);
      psum += ops;
      pcnt += opc;
      if (omv < mv || (omv == mv && omi < mi)) { mv = omv; mi = omi; }
    }
    if (lane == 0) {
      const int p = blockIdx.x;
      ws_m[p] = m; ws_s[p] = s; ws_ps[p] = psum;
      ws_pc[p] = pcnt; ws_mv[p] = mv; ws_mi[p] = mi;
    }
  }
}

// ---------------------------------------------------------------------------
// Kernel 2: combine TEAM partials per anchor -> per_sample; copy hard_pos row
// ---------------------------------------------------------------------------
__global__ __launch_bounds__(NTHREADS) void wscl_combine(
    const float* __restrict__ mem,       // [N, D]
    const float* __restrict__ ws_m,  const float* __restrict__ ws_s,
    const float* __restrict__ ws_ps, const float* __restrict__ ws_pc,
    const float* __restrict__ ws_mv, const int* __restrict__ ws_mi,
    float* __restrict__ per_sample,      // [B]
    float* __restrict__ out_hard,        // [B, D]
    int D) {
  const int b = blockIdx.x;
  __shared__ int sh_idx;
  if (threadIdx.x == 0) {
    float M = NEG_SENTINEL, S = 0.f, PS = 0.f, PC = 0.f;
    float MV = FLT_MAX;
    int   MI = 0x7fffffff;
    for (int sp = 0; sp < TEAM; ++sp) {
      const int p = b * TEAM + sp;
      lse_merge(M, S, ws_m[p], ws_s[p]);
      PS += ws_ps[p];
      PC += ws_pc[p];
      const float mv = ws_mv[p];
      const int   mi = ws_mi[p];
      if (mv < MV || (mv == MV && mi < MI)) { MV = mv; MI = mi; }
    }
    // mean_log_prob_pos = (sum(adc_pos) - PC*mx - PC*log(sum_exp)) / PC
    const float mlpp = (PS - PC * M - PC * __logf(S)) / PC;
    per_sample[b] = -(TEMPERATURE / BASE_TEMPERATURE) * mlpp;
    sh_idx = MI;
    __builtin_prefetch(mem + (size_t)MI * (size_t)D, 0, 0);  // global_prefetch_b8
  }
  __syncthreads();
  const int hidx = sh_idx;
  const float* src = mem + (size_t)hidx * (size_t)D;
  float* dst = out_hard + (size_t)b * (size_t)D;
  for (int i = threadIdx.x; i < D; i += blockDim.x) dst[i] = src[i];
}

// ---------------------------------------------------------------------------
// Kernel 3: loss = mean(per_sample)
// ---------------------------------------------------------------------------
__global__ __launch_bounds__(256) void wscl_loss(
    const float* __restrict__ per_sample, float* __restrict__ out, int B) {
  __shared__ float red[256];
  float acc = 0.f;
  for (int i = threadIdx.x; i < B; i += 256) acc += per_sample[i];
  red[threadIdx.x] = acc;
  __syncthreads();
  for (int off = 128; off > 0; off >>= 1) {
    if ((int)threadIdx.x < off) red[threadIdx.x] += red[threadIdx.x + off];
    __syncthreads();
  }
  if (threadIdx.x == 0) out[0] = red[0] / (float)B;
}

// ---------------------------------------------------------------------------
extern "C" void kernel_launch(void* const* d_in, const int* in_sizes, int n_in,
                              void* d_out, int out_size, void* d_ws, size_t ws_size,
                              hipStream_t stream) {
  const float* mem      = (const float*)d_in[0];  // [N, D]
  const float* logits   = (const float*)d_in[1];  // [B, N]
  const int*   memCID   = (const int*)d_in[2];    // [N]
  const int*   memTID   = (const int*)d_in[3];    // [N]
  const int*   camids   = (const int*)d_in[4];    // [B]
  const int*   trackids = (const int*)d_in[5];    // [B]

  const int N = in_sizes[2];
  const int B = in_sizes[4];
  const int D = in_sizes[0] / N;

  float* out      = (float*)d_out;
  float* out_loss = out;        // element 0
  float* out_hard = out + 1;    // [B, D]

  const int P = B * TEAM;
  float* wsf        = (float*)d_ws;
  float* ws_m       = wsf + 0 * P;
  float* ws_s       = wsf + 1 * P;
  float* ws_ps      = wsf + 2 * P;
  float* ws_pc      = wsf + 3 * P;
  float* ws_mv      = wsf + 4 * P;
  float* per_sample = wsf + 5 * P;
  int*   ws_mi      = (int*)(per_sample + B);

  wscl_partial<<<dim3(B * TEAM), dim3(NTHREADS), 0, stream>>>(
      logits, memCID, memTID, camids, trackids, N,
      ws_m, ws_s, ws_ps, ws_pc, ws_mv, ws_mi);

  wscl_combine<<<dim3(B), dim3(NTHREADS), 0, stream>>>(
      mem, ws_m, ws_s, ws_ps, ws_pc, ws_mv, ws_mi, per_sample, out_hard, D);

  wscl_loss<<<dim3(1), dim3(256), 0, stream>>>(per_sample, out_loss, B);
}